// NegLogLikelihood_13099650253073
// MI455X (gfx1250) — compile-verified
//
#include <hip/hip_runtime.h>
#include <hip/hip_bf16.h>

// NLL over p = pred[:, :, 0]:  out = -sum( t ? log(p) : log1p(-p) ) / BATCH
//
// MI455X reasoning:
//  * pred is 128 MB; channel-0 stride = 32 B, so every 128 B HBM line is
//    touched regardless -> hard floor ~132 MB / 23.3 TB/s ~= 5.7 us. Pure
//    streaming reduce; one v_log_f32 per point (select folded into the log).
//  * Each lane owns one 128 B line of pred per step (4 points) + 4 packed
//    target bytes; global_prefetch_b8 one grid-stride ahead (working set
//    fits the 192 MB L2, so graph replays can run from L2).
//  * Final reduction of block partials uses V_WMMA_F32_16X16X4_F32 (f32
//    in/out -> no precision loss) with an all-ones B matrix: D = A*1 + C
//    sums 64 floats per issue on a single wave32 (EXEC all ones).

typedef __attribute__((ext_vector_type(2))) float v2f;
typedef __attribute__((ext_vector_type(8))) float v8f;

#define NLL_BATCH   256
#define NLL_GRIDS   16384
#define NLL_TYPES   8
#define NLL_POINTS  (NLL_BATCH * NLL_GRIDS)   // 4,194,304

#define BLOCKS_A    2048
#define THREADS_A   256                        // 8 waves per WGP-dispatch unit

__global__ __launch_bounds__(THREADS_A)
void nll_partial_kernel(const float* __restrict__ pred,
                        const unsigned char* __restrict__ tgt,
                        float* __restrict__ partials)
{
    const int tid      = threadIdx.x;
    const int gid      = blockIdx.x * THREADS_A + tid;
    const int nthreads = gridDim.x * THREADS_A;
    const int step     = nthreads * 4;           // points per grid sweep

    const unsigned int* tgt4 = (const unsigned int*)tgt;  // 4 bools per load

    float acc = 0.0f;
    for (int base = gid * 4; base < NLL_POINTS; base += step) {
        // prefetch next sweep's line for this lane (speculative: OOB at the
        // tail is silently dropped per ISA prefetch rules)
        __builtin_prefetch(&pred[(size_t)(base + step) * NLL_TYPES], 0, 0);

        unsigned int t4 = tgt4[base >> 2];
#pragma unroll
        for (int j = 0; j < 4; ++j) {
            // channel-0 of point (base+j): offsets +0/+32/+64/+96 B => this
            // lane consumes exactly one 128 B line of pred per outer step
            float p = pred[(size_t)(base + j) * NLL_TYPES];
            bool  t = (t4 >> (8 * j)) & 1u;
            float q = t ? p : (1.0f - p);      // log1p(-p) ~= log(1-p), p>=1e-4
            acc += __logf(q);                  // one v_log_f32 per point
        }
    }

    // block tree-reduce (deterministic)
    __shared__ float smem[THREADS_A];
    smem[tid] = acc;
    __syncthreads();
#pragma unroll
    for (int off = THREADS_A / 2; off > 0; off >>= 1) {
        if (tid < off) smem[tid] += smem[tid + off];
        __syncthreads();
    }
    if (tid == 0) partials[blockIdx.x] = smem[0];
}

// Single wave32, EXEC all ones (WMMA requirement). Sums BLOCKS_A partials
// 64-at-a-time through V_WMMA_F32_16X16X4_F32 with B == ones:
//   D[m][n] = rowsum(A[m][:]) + C[m][n]   (every column identical)
// After the loop, lane L<16 holds rows 0..7 in c[0..7] (col N=L), lanes 16..31
// hold rows 8..15, so  total = colsum(lane 0) + colsum(lane 16).
__global__ __launch_bounds__(32)
void nll_final_kernel(const float* __restrict__ partials,
                      float* __restrict__ out)
{
    const int lane = threadIdx.x;

    v8f c = {};
    v2f ones;
    ones.x = 1.0f;
    ones.y = 1.0f;

#pragma unroll
    for (int i = 0; i < BLOCKS_A / 64; ++i) {
        v2f a;
        a.x = partials[i * 64 + lane * 2 + 0];
        a.y = partials[i * 64 + lane * 2 + 1];
        // 8 args: (neg_a, A, neg_b, B, c_mod, C, reuse_a, reuse_b)
        c = __builtin_amdgcn_wmma_f32_16x16x4_f32(
                /*neg_a=*/false, a, /*neg_b=*/false, ones,
                /*c_mod=*/(short)0, c, /*reuse_a=*/false, /*reuse_b=*/false);
    }

    float s = c[0] + c[1] + c[2] + c[3] + c[4] + c[5] + c[6] + c[7];
    float total = __shfl(s, 0, 32) + __shfl(s, 16, 32);

    if (lane == 0) out[0] = -total / (float)NLL_BATCH;
}

extern "C" void kernel_launch(void* const* d_in, const int* in_sizes, int n_in,
                              void* d_out, int out_size, void* d_ws, size_t ws_size,
                              hipStream_t stream)
{
    const float*         pred = (const float*)d_in[0];          // (256,16384,8) f32
    const unsigned char* tgt  = (const unsigned char*)d_in[1];  // (256,16384) bool (1B)
    float* partials = (float*)d_ws;                             // BLOCKS_A floats
    float* out      = (float*)d_out;                            // scalar f32

    nll_partial_kernel<<<BLOCKS_A, THREADS_A, 0, stream>>>(pred, tgt, partials);
    nll_final_kernel<<<1, 32, 0, stream>>>(partials, out);
}